// OneDilate_1580547972907
// MI455X (gfx1250) — compile-verified
//
#include <hip/hip_runtime.h>

// CDNA5 / gfx1250: 10x10 all-ones depthwise box filter on (1-x)*0.5 with
// edge-replicate padding, as two banded-ones matrix multiplies on the
// WMMA f32 16x16x4 pipe.  Memory-bound (~200MB floor @ 23.3TB/s): a 64x64
// output superblock per workgroup shares one 73x73 halo in LDS (1.30x
// read amplification), 16 waves each run the 21-WMMA separable box-sum
// on their 16x16 quadrant tile.

typedef float v2f __attribute__((ext_vector_type(2)));
typedef float v8f __attribute__((ext_vector_type(8)));

#define IMG_H 512
#define IMG_W 512
#define OUT_H 511
#define OUT_W 511
#define PAD   4            // (10-1)/2
#define SB    64           // superblock output size (per dim)
#define SBS   8            // superblocks per dim (8*64 = 512 >= 511)
#define WAVES_PER_BLOCK 16 // 4x4 quadrant tiles of 16x16

// Shared input tile: rows 0..79 (need 73 + stage-H over-read), cols 0..75
#define Y_ROWS 80
#define Y_COLS 77          // stride pad (76 used); 77 co-prime-ish with 64 banks
#define Y_ELEMS (Y_ROWS * (Y_COLS - 1))   // 80*76 elements actually filled

// Per-wave intermediate U (horizontal sums), D-layout -> B-layout bounce
#define U_ROWS 28
#define U_COLS 17          // stride pad (16 used)

__global__ __launch_bounds__(32 * WAVES_PER_BLOCK)
void box10_wmma_kernel(const float* __restrict__ x, float* __restrict__ out) {
    __shared__ float sY[Y_ROWS * Y_COLS];
    __shared__ float sU[WAVES_PER_BLOCK][U_ROWS * U_COLS];

    const int tid  = threadIdx.x;
    const int lane = tid & 31;
    const int wave = tid >> 5;
    const int hsel = lane >> 4;      // 0: lanes 0-15, 1: lanes 16-31
    const int lid  = lane & 15;

    float* U = sU[wave];

    // block id -> (image, superblock row, superblock col)
    const int sbc = blockIdx.x & (SBS - 1);
    const int sbr = (blockIdx.x >> 3) & (SBS - 1);
    const int img = blockIdx.x >> 6;                 // 0..95 (=32*3)

    const float* __restrict__ xin = x   + (size_t)img * IMG_H * IMG_W;
    float*       __restrict__ op  = out + (size_t)img * OUT_H * OUT_W;

    const int base_r = sbr * SB - PAD;               // padded-space origin
    const int base_c = sbc * SB - PAD;

    // ---- Cooperative fill: 80 rows x 76 cols, edge-clamped, y=(1-x)*0.5 ----
    #pragma unroll
    for (int t = 0; t < 12; ++t) {                   // 12*512 >= 6080
        const int idx = t * (32 * WAVES_PER_BLOCK) + tid;
        if (idx < Y_ELEMS) {
            const int rr = idx / (Y_COLS - 1);
            const int cc = idx % (Y_COLS - 1);
            int gr = base_r + rr; gr = gr < 0 ? 0 : (gr > IMG_H - 1 ? IMG_H - 1 : gr);
            int gc = base_c + cc; gc = gc < 0 ? 0 : (gc > IMG_W - 1 ? IMG_W - 1 : gc);
            const float v = xin[gr * IMG_W + gc];
            sY[rr * Y_COLS + cc] = (1.0f - v) * 0.5f;
        }
    }
    __syncthreads();

    // Quadrant tile of this wave inside the superblock: 4x4 grid of 16x16
    const int qr = wave >> 2;
    const int qc = wave & 3;
    const float* Yq = sY + (qr * 16) * Y_COLS + qc * 16;  // wave's halo view

    // ---- Stage H:  U = Y_tile x Sh   (Sh[q][c] = 1 iff q-9 <= c <= q) ----
    #pragma unroll
    for (int rb = 0; rb < 2; ++rb) {
        v8f acc = {};
        #pragma unroll
        for (int k = 0; k < 28; k += 4) {
            const int kk = k + 2 * hsel;
            // A operand: Y rows (M=lid within block), K-pair per lane half
            v2f A;
            A.x = Yq[(rb * 16 + lid) * Y_COLS + kk];
            A.y = Yq[(rb * 16 + lid) * Y_COLS + kk + 1];
            // B operand: banded Sh generated in registers
            v2f B;
            B.x = (lid >= kk - 9 && lid <= kk)     ? 1.0f : 0.0f;
            B.y = (lid >= kk - 8 && lid <= kk + 1) ? 1.0f : 0.0f;
            acc = __builtin_amdgcn_wmma_f32_16x16x4_f32(
                      false, A, false, B, (short)0, acc, false, false);
        }
        // D layout -> LDS: VGPR j holds row (rb*16 + j + 8*hsel), col lid
        #pragma unroll
        for (int j = 0; j < 8; ++j) {
            const int ur = rb * 16 + j + 8 * hsel;
            if (ur < U_ROWS) U[ur * U_COLS + lid] = acc[j];
        }
    }
    __syncthreads();

    // ---- Stage V:  Out = Sv x U   (Sv[r][m] = 1 iff r <= m <= r+9) ----
    v8f acc = {};
    #pragma unroll
    for (int k = 0; k < 28; k += 4) {
        const int kk = k + 2 * hsel;
        v2f A;                                  // banded Sv in registers
        A.x = (kk     >= lid && kk     <= lid + 9) ? 1.0f : 0.0f;
        A.y = (kk + 1 >= lid && kk + 1 <= lid + 9) ? 1.0f : 0.0f;
        v2f B;                                  // U rows from LDS (B layout)
        B.x = U[kk * U_COLS + lid];
        B.y = U[(kk + 1) * U_COLS + lid];
        acc = __builtin_amdgcn_wmma_f32_16x16x4_f32(
                  false, A, false, B, (short)0, acc, false, false);
    }

    // ---- Store 16x16 output tile from D layout, bounds-checked ----
    const int oc = sbc * SB + qc * 16 + lid;
    const int or0 = sbr * SB + qr * 16;
    #pragma unroll
    for (int j = 0; j < 8; ++j) {
        const int orow = or0 + j + 8 * hsel;
        if (orow < OUT_H && oc < OUT_W)
            op[(size_t)orow * OUT_W + oc] = acc[j];
    }
}

extern "C" void kernel_launch(void* const* d_in, const int* in_sizes, int n_in,
                              void* d_out, int out_size, void* d_ws, size_t ws_size,
                              hipStream_t stream) {
    (void)in_sizes; (void)n_in; (void)d_ws; (void)ws_size; (void)out_size;
    const float* x = (const float*)d_in[0];
    float* out = (float*)d_out;

    // 32 batch * 3 ch = 96 images, 8x8 superblocks each
    const int blocks = 96 * SBS * SBS;               // 6144
    box10_wmma_kernel<<<blocks, 32 * WAVES_PER_BLOCK, 0, stream>>>(x, out);
}